// OHEM_16776142259012
// MI455X (gfx1250) — compile-verified
//
#include <hip/hip_runtime.h>
#include <stdint.h>

// Problem geometry: x,y are [32,2,768,768] fp32 -> 64 rows of 589824 elements.
#define ROWS   64
#define NTOT   589824
#define PB     36            // blocks per row
#define EPB    16384         // elements per block (36*16384 = 589824 exactly)
#define CHUNK  1024          // elements per async chunk (256 lanes * 4 floats)
#define NCH    16            // chunks per block
#define THRV   0.01f

// Workspace layout in 32-bit words (~1.01 MB total)
#define H1OFF  0             // 64 * 2048 bins (bits[30:20])
#define H2OFF  131072        // 64 * 1024 bins (bits[19:10])
#define H3OFF  196608        // 64 * 1024 bins (bits[ 9: 0])
#define NPOSO  262144        // 64
#define B1O    262208        // 64
#define K1O    262272        // 64
#define KEY2O  262336        // 64
#define K2O    262400        // 64
#define CGTO   262464        // 64 : count strictly greater than tau
#define NHEO   262528        // 64
#define TAUO   262592        // 64 : tau as raw float bits
#define DONEO  262656        // 64 : "take all negatives" flag
#define PSO    262720        // 2304 per-block partial sums (float)
#define WSWORDS 265024

// ---------------- CDNA5 async copy to LDS (ASYNCcnt path) ----------------
__device__ __forceinline__ void async_cp16(uint32_t lds_off, const void* gptr) {
  // GLOBAL_LOAD_ASYNC_TO_LDS_B128, GV mode: vdst = per-lane LDS byte offset,
  // vaddr = 64-bit global address, saddr = off. Tracked on ASYNCcnt.
  asm volatile("global_load_async_to_lds_b128 %0, %1, off"
               :: "v"(lds_off), "v"(gptr) : "memory");
}
__device__ __forceinline__ void async_wait0() {
  asm volatile("s_wait_asynccnt 0" ::: "memory");
}
__device__ __forceinline__ void async_wait2() {
  asm volatile("s_wait_asynccnt 2" ::: "memory");
}

// ---------------- elementwise BCE-with-logits ----------------
__device__ __forceinline__ float bce_loss(float xv, float yv) {
  // softplus(x) = max(x,0) + log1p(exp(-|x|))  (stable, matches jnp.logaddexp(x,0))
  float sp = fmaxf(xv, 0.0f) + log1pf(__expf(-fabsf(xv)));
  return sp - xv * yv;
}

// ---------------- WMMA-based wave32 sum reduction ----------------
typedef float v2f __attribute__((ext_vector_type(2)));
typedef float v8f __attribute__((ext_vector_type(8)));

__device__ __forceinline__ float wave_sum(float p) {
#if __has_builtin(__builtin_amdgcn_wmma_f32_16x16x4_f32)
  // A (16x4): a0 = p (lane m / m+16 supply rows), a1 = 0. B = ones (4x16).
  // D[m][n] = p_m + p_{m+16}.  Lanes 0-15 hold rows 0..7 in c[0..7],
  // lanes 16-31 hold rows 8..15.  In-lane 8-add + SWAPX16 = full wave sum.
  v2f a; a[0] = p;    a[1] = 0.0f;
  v2f b; b[0] = 1.0f; b[1] = 1.0f;
  v8f c = {0.f,0.f,0.f,0.f,0.f,0.f,0.f,0.f};
  c = __builtin_amdgcn_wmma_f32_16x16x4_f32(false, a, false, b, (short)0, c, false, false);
  float s = c[0]+c[1]+c[2]+c[3]+c[4]+c[5]+c[6]+c[7];
  float o = __shfl_xor(s, 16, 32);
  return s + o;
#else
  float s = p;
  for (int m = 16; m >= 1; m >>= 1) s += __shfl_xor(s, m, 32);
  return s;
#endif
}

// ---------------- streaming passes ----------------
// MODE 0: level-1 histogram (bits>>20) + npos count
// MODE 1: level-2 histogram ((bits>>10)&1023) restricted to bin1 == b1[row]
// MODE 2: level-3 histogram (bits&1023)     restricted to bits>>10 == key2[row]
// MODE 3: sum of loss where bits > tau[row]  (per-block partial, deterministic)
template <int MODE>
__global__ void __launch_bounds__(256)
pass_k(const float* __restrict__ x, const float* __restrict__ y,
       uint32_t* __restrict__ ws) {
  constexpr int NB = (MODE == 0) ? 2048 : ((MODE == 3) ? 1 : 1024);
  __shared__ uint32_t lh[NB];
  __shared__ __align__(16) float bufx[2][CHUNK];
  __shared__ __align__(16) float bufy[2][CHUNK];
  __shared__ uint32_t s_np;
  __shared__ float s_wsum[8];

  const int t    = threadIdx.x;
  const int row  = blockIdx.x / PB;
  const int slab = blockIdx.x - row * PB;
  const size_t base = (size_t)row * NTOT + (size_t)slab * EPB;

  uint32_t mb1 = 0, mkey2 = 0, mtau = 0;
  if (MODE == 1) { mb1 = ws[B1O + row]; if (mb1 > 2047u) return; }     // done row: skip
  if (MODE == 2) { if (ws[DONEO + row]) return; mkey2 = ws[KEY2O + row]; }
  if (MODE == 3) { mtau = ws[TAUO + row]; }

  if (MODE != 3) { for (int i = t; i < NB; i += 256) lh[i] = 0; }
  if (MODE == 0 && t == 0) s_np = 0;
  __syncthreads();

  // Per-lane LDS byte offsets (generic LDS pointer's low 32 bits = LDS address)
  uint32_t lx[2], ly[2];
  lx[0] = (uint32_t)(uintptr_t)&bufx[0][t * 4];
  lx[1] = (uint32_t)(uintptr_t)&bufx[1][t * 4];
  ly[0] = (uint32_t)(uintptr_t)&bufy[0][t * 4];
  ly[1] = (uint32_t)(uintptr_t)&bufy[1][t * 4];

  // prologue: async-stage chunk 0
  async_cp16(lx[0], x + base + (size_t)t * 4);
  async_cp16(ly[0], y + base + (size_t)t * 4);

  uint32_t npc = 0;
  float lsum = 0.0f;

  for (int c = 0; c < NCH; ++c) {
    const int cur = c & 1;
    if (c + 1 < NCH) {
      const size_t nb = base + (size_t)(c + 1) * CHUNK + (size_t)t * 4;
      async_cp16(lx[1 - cur], x + nb);
      async_cp16(ly[1 - cur], y + nb);
      async_wait2();                  // current chunk's 2 ops complete (in-order)
    } else {
      async_wait0();
    }
    float4 xv = *reinterpret_cast<const float4*>(&bufx[cur][t * 4]);
    float4 yv = *reinterpret_cast<const float4*>(&bufy[cur][t * 4]);
    const float xs[4] = {xv.x, xv.y, xv.z, xv.w};
    const float ys[4] = {yv.x, yv.y, yv.z, yv.w};
#pragma unroll
    for (int j = 0; j < 4; ++j) {
      float l = bce_loss(xs[j], ys[j]);
      if (MODE == 0) {
        if (l < THRV) npc++;
        else atomicAdd(&lh[__float_as_uint(l) >> 20], 1u);
      } else if (MODE == 1) {
        if (l >= THRV) {
          uint32_t b = __float_as_uint(l);
          if ((b >> 20) == mb1) atomicAdd(&lh[(b >> 10) & 1023u], 1u);
        }
      } else if (MODE == 2) {
        if (l >= THRV) {
          uint32_t b = __float_as_uint(l);
          if ((b >> 10) == mkey2) atomicAdd(&lh[b & 1023u], 1u);
        }
      } else {
        if (l >= THRV) {
          uint32_t b = __float_as_uint(l);
          if (b > mtau) lsum += l;
        }
      }
    }
  }

  if (MODE == 3) {
    float tot = wave_sum(lsum);               // EXEC all-ones here (no divergence)
    if ((t & 31) == 0) s_wsum[t >> 5] = tot;
    __syncthreads();
    if (t == 0) {
      float bs = 0.f;
      for (int w = 0; w < 8; ++w) bs += s_wsum[w];
      ((float*)ws)[PSO + blockIdx.x] = bs;    // deterministic per-block slot
    }
    return;
  }

  if (MODE == 0) atomicAdd(&s_np, npc);
  __syncthreads();
  uint32_t* gh = ws + ((MODE == 0) ? (H1OFF + row * 2048)
                     : (MODE == 1) ? (H2OFF + row * 1024)
                                   : (H3OFF + row * 1024));
  for (int i = t; i < NB; i += 256) {
    uint32_t v = lh[i];
    if (v) atomicAdd(&gh[i], v);
  }
  if (MODE == 0 && t == 0) atomicAdd(&ws[NPOSO + row], s_np);
}

// ---------------- top-k bin selection (descending suffix scan) ----------------
__device__ void select_topk_n(const uint32_t* __restrict__ hist, int nb, uint32_t k,
                              uint32_t* bin_out, uint32_t* above_out) {
  __shared__ uint32_t part[256];
  __shared__ uint32_t s_bin, s_abv;
  const int bpt = nb / 256;
  const int t = threadIdx.x;
  uint32_t s = 0;
  for (int i = 0; i < bpt; ++i) s += hist[t * bpt + i];
  part[t] = s;
  __syncthreads();
  if (t == 0) {
    uint32_t above = 0;
    int ti;
    for (ti = 255; ti > 0; --ti) {
      if (above + part[ti] >= k) break;
      above += part[ti];
    }
    int b;
    for (b = ti * bpt + bpt - 1; b > ti * bpt; --b) {
      uint32_t h = hist[b];
      if (above + h >= k) break;
      above += h;
    }
    s_bin = (uint32_t)b;
    s_abv = above;
  }
  __syncthreads();
  *bin_out = s_bin;
  *above_out = s_abv;
}

__global__ void __launch_bounds__(256) meta1_k(uint32_t* __restrict__ ws) {
  const int row = blockIdx.x;
  uint32_t npos = ws[NPOSO + row];
  uint32_t nneg = (uint32_t)NTOT - npos;
  float nhef = fminf((float)nneg, 3.0f * (float)npos);
  nhef = fmaxf(nhef, 0.005f * (float)NTOT);
  uint32_t nhe = (uint32_t)ceilf(nhef);
  if (nhe < 1u) nhe = 1u;
  if (nhe >= nneg) {        // keep every negative: tau = +0.0, tie term zeroed
    if (threadIdx.x == 0) {
      ws[DONEO + row] = 1u;
      ws[NHEO + row]  = nhe;
      ws[TAUO + row]  = 0u;
      ws[CGTO + row]  = nhe;
      ws[B1O + row]   = 0xFFFFFFFFu;
    }
    return;
  }
  uint32_t b, a;
  select_topk_n(ws + H1OFF + row * 2048, 2048, nhe, &b, &a);
  if (threadIdx.x == 0) {
    ws[DONEO + row] = 0u;
    ws[NHEO + row]  = nhe;
    ws[B1O + row]   = b;
    ws[K1O + row]   = nhe - a;
    ws[CGTO + row]  = a;
  }
}

__global__ void __launch_bounds__(256) meta2_k(uint32_t* __restrict__ ws) {
  const int row = blockIdx.x;
  if (ws[DONEO + row]) return;
  uint32_t b, a;
  select_topk_n(ws + H2OFF + row * 1024, 1024, ws[K1O + row], &b, &a);
  if (threadIdx.x == 0) {
    ws[KEY2O + row] = (ws[B1O + row] << 10) | b;
    ws[K2O + row]   = ws[K1O + row] - a;
    ws[CGTO + row] += a;
  }
}

__global__ void __launch_bounds__(256) meta3_k(uint32_t* __restrict__ ws) {
  const int row = blockIdx.x;
  if (ws[DONEO + row]) return;
  uint32_t b, a;
  select_topk_n(ws + H3OFF + row * 1024, 1024, ws[K2O + row], &b, &a);
  if (threadIdx.x == 0) {
    ws[TAUO + row]  = (ws[KEY2O + row] << 10) | b;   // exact 31-bit tau pattern
    ws[CGTO + row] += a;
  }
}

__global__ void __launch_bounds__(64) final_k(const uint32_t* __restrict__ ws,
                                              float* __restrict__ out) {
  __shared__ float acc[ROWS];
  const int r = threadIdx.x;
  const float* ps = (const float*)(ws + PSO);
  float s = 0.f;
  for (int i = 0; i < PB; ++i) s += ps[r * PB + i];   // deterministic order
  float nhe  = (float)ws[NHEO + r];
  float tau  = __uint_as_float(ws[TAUO + r]);
  float ties = (float)(ws[NHEO + r] - ws[CGTO + r]);
  acc[r] = (s + ties * tau) / nhe;
  __syncthreads();
  if (r == 0) {
    float m = 0.f;
    for (int i = 0; i < ROWS; ++i) m += acc[i];
    out[0] = m / (float)ROWS;
  }
}

extern "C" void kernel_launch(void* const* d_in, const int* in_sizes, int n_in,
                              void* d_out, int out_size, void* d_ws, size_t ws_size,
                              hipStream_t stream) {
  (void)in_sizes; (void)n_in; (void)out_size; (void)ws_size;
  const float* x = (const float*)d_in[0];
  const float* y = (const float*)d_in[1];
  uint32_t* ws = (uint32_t*)d_ws;

  hipMemsetAsync(d_ws, 0, (size_t)WSWORDS * 4u, stream);

  dim3 bt(256);
  pass_k<0><<<ROWS * PB, bt, 0, stream>>>(x, y, ws);
  meta1_k<<<ROWS, bt, 0, stream>>>(ws);
  pass_k<1><<<ROWS * PB, bt, 0, stream>>>(x, y, ws);
  meta2_k<<<ROWS, bt, 0, stream>>>(ws);
  pass_k<2><<<ROWS * PB, bt, 0, stream>>>(x, y, ws);
  meta3_k<<<ROWS, bt, 0, stream>>>(ws);
  pass_k<3><<<ROWS * PB, bt, 0, stream>>>(x, y, ws);
  final_k<<<1, ROWS, 0, stream>>>(ws, (float*)d_out);
}